// Painting_20538533610174
// MI455X (gfx1250) — compile-verified
//
#include <hip/hip_runtime.h>
#include <cstdint>

// Composite 256 RGBA strokes over a background canvas (alpha-over, scan order).
// out[c][p] = scan_i ( canvas*(1-a_i) + a_i*stroke_i ), canvas alpha init = 1.
//
// Memory-bound streaming kernel: 1.08 GB single-use traffic @ 23.3 TB/s
// => ~46 us floor. gfx1250 async global->LDS DMA (global_load_async_to_lds_b128,
// NT-hinted, clause-grouped) forms a DEPTH-deep per-wave staging FIFO throttled
// with s_wait_asynccnt; blend consumes via ds_load_b128.

#define N_STROKES 256
#define HW_PIX    (512 * 512)
#define THREADS   128
#define PPT       4                        // pixels per thread (one float4)
#define TILE_PIX  (THREADS * PPT)          // 512 pixels per block
#define DEPTH     8                        // async pipeline stages (64 KB LDS)
#define STAGE_F4  (4 * THREADS)            // float4 slots per stage (4 channels)
#define CH_LDS_B  (THREADS * 16)           // bytes per channel within a stage
#define CH_MEM_B  (HW_PIX * 4)             // bytes per channel plane in HBM

// Native clang vector type: required by __builtin_nontemporal_store (HIP's
// float4 is a struct and is rejected). Same 16-byte layout.
typedef float vf4 __attribute__((ext_vector_type(4)));

__device__ __forceinline__ float4 blend4(float4 c, float4 s, float4 a) {
  // c + a*(s - c)  == c*(1-a) + a*s   (2 VALU ops per component)
  c.x = fmaf(a.x, s.x - c.x, c.x);
  c.y = fmaf(a.y, s.y - c.y, c.y);
  c.z = fmaf(a.z, s.z - c.z, c.z);
  c.w = fmaf(a.w, s.w - c.w, c.w);
  return c;
}

__device__ __forceinline__ void nt_store4(float* p, float4 v) {
  vf4 x;
  x.x = v.x; x.y = v.y; x.z = v.z; x.w = v.w;
  __builtin_nontemporal_store(x, (vf4*)p);
}

__global__ __launch_bounds__(THREADS) void composite_kernel(
    const float* __restrict__ strokes,   // [256][4][512][512]
    const float* __restrict__ bg,        // [1][4][512][512]
    float* __restrict__ out)             // [1][4][512][512]
{
  __shared__ float4 stage[DEPTH * STAGE_F4];   // 64 KB FIFO: [stage][ch][tid]

  const int tid = threadIdx.x;
  const int pix = blockIdx.x * TILE_PIX + tid * PPT;   // 4 consecutive pixels

  // Low 32 bits of a generic pointer into LDS == LDS byte offset (aperture layout).
  const unsigned ldsBase =
      (unsigned)(uintptr_t)(&stage[0]) + (unsigned)(tid * sizeof(float4));
  const unsigned pixByte = (unsigned)pix * 4u;

  // Init canvas from background; canvas alpha channel forced to 1.0.
  float4 canR = *(const float4*)(bg + 0 * HW_PIX + pix);
  float4 canG = *(const float4*)(bg + 1 * HW_PIX + pix);
  float4 canB = *(const float4*)(bg + 2 * HW_PIX + pix);
  float4 canA = make_float4(1.f, 1.f, 1.f, 1.f);

  // Issue one stroke tile (4 channels x 16B/lane) into its FIFO slot as a
  // clause-grouped burst of NT async DMAs (zero-reuse stream: keep L2 clean).
  auto issue_stage = [&](int stroke) {
    const int slot = stroke & (DEPTH - 1);
    const unsigned l0 = ldsBase + (unsigned)(slot * STAGE_F4 * (int)sizeof(float4));
    const unsigned g0 = (unsigned)stroke * (4u * (unsigned)HW_PIX * 4u) + pixByte;
    asm volatile(
        "s_clause 0x3\n\t"
        "global_load_async_to_lds_b128 %0, %4, %8 th:TH_LOAD_NT\n\t"
        "global_load_async_to_lds_b128 %1, %5, %8 th:TH_LOAD_NT\n\t"
        "global_load_async_to_lds_b128 %2, %6, %8 th:TH_LOAD_NT\n\t"
        "global_load_async_to_lds_b128 %3, %7, %8 th:TH_LOAD_NT"
        :: "v"(l0),                "v"(l0 + 1 * CH_LDS_B),
           "v"(l0 + 2 * CH_LDS_B), "v"(l0 + 3 * CH_LDS_B),
           "v"(g0),                "v"(g0 + 1 * CH_MEM_B),
           "v"(g0 + 2 * CH_MEM_B), "v"(g0 + 3 * CH_MEM_B),
           "s"(strokes)
        : "memory");
  };

  // Prologue: fill the pipeline (DEPTH stages * 4 instructions, ASYNCcnt <= 32).
#pragma unroll
  for (int s = 0; s < DEPTH; ++s) issue_stage(s);

  for (int i = 0; i < N_STROKES; ++i) {
    // 4 instructions per stage; async loads retire in order, so waiting for
    // <= 4*(DEPTH-1) outstanding guarantees the oldest stage has landed.
    asm volatile("s_wait_asynccnt %0" :: "n"(4 * (DEPTH - 1)) : "memory");

    const int slot = i & (DEPTH - 1);
    const float4 sR = stage[(slot * 4 + 0) * THREADS + tid];
    const float4 sG = stage[(slot * 4 + 1) * THREADS + tid];
    const float4 sB = stage[(slot * 4 + 2) * THREADS + tid];
    const float4 sA = stage[(slot * 4 + 3) * THREADS + tid];

    canR = blend4(canR, sR, sA);
    canG = blend4(canG, sG, sA);
    canB = blend4(canB, sB, sA);
    canA = blend4(canA, sA, sA);

    // WAR guard: LDS reads of this slot must complete before the async DMA
    // below is allowed to overwrite it.
    asm volatile("s_wait_dscnt 0x0" ::: "memory");

    // Refill the slot we just consumed. Tail iterations wrap (dummy refetch of
    // strokes 0..DEPTH-1, never consumed) so the wait immediate stays constant.
    issue_stage((i + DEPTH) & (N_STROKES - 1));
  }

  // Single-use output: NT store, rinse through to HBM without occupying L2.
  nt_store4(out + 0 * HW_PIX + pix, canR);
  nt_store4(out + 1 * HW_PIX + pix, canG);
  nt_store4(out + 2 * HW_PIX + pix, canB);
  nt_store4(out + 3 * HW_PIX + pix, canA);
}

extern "C" void kernel_launch(void* const* d_in, const int* in_sizes, int n_in,
                              void* d_out, int out_size, void* d_ws, size_t ws_size,
                              hipStream_t stream) {
  (void)in_sizes; (void)n_in; (void)out_size; (void)d_ws; (void)ws_size;
  const float* strokes = (const float*)d_in[0];   // [256,4,512,512] f32
  const float* bg      = (const float*)d_in[1];   // [1,4,512,512]   f32
  float* out           = (float*)d_out;           // [1,4,512,512]   f32
  composite_kernel<<<dim3(HW_PIX / TILE_PIX), dim3(THREADS), 0, stream>>>(
      strokes, bg, out);
}